// Rank_CNN_46926812676740
// MI455X (gfx1250) — compile-verified
//
#include <hip/hip_runtime.h>
#include <hip/hip_bf16.h>

// ---------------------------------------------------------------------------
// Problem constants (from reference)
// ---------------------------------------------------------------------------
#define NN        8000      // samples
#define TT        64        // time steps
#define FF        64        // raw features
#define EE        8         // stat features
#define WIN       40        // window
#define KER       5         // conv kernel
#define INCH      144       // F + 10*E
#define LOUT      36        // WIN + 1 - KER
#define ALPHA     0.01f

// Stats buffer layout per (n, tau): 80 floats
//  0..7  wmean | 8..15 wstd | 16..23 rank | 24..31 wmax | 32..39 wmin
// 40..47 mmean |48..55 mstd | 56..63 rank | 64..71 mmax | 72..79 mmin
#define SDIM      80

// Workspace layout (floats)
#define S_OFF     0
#define S_ELEMS   (NN * WIN * SDIM)           // 25,600,000
#define L_OFF     (S_OFF + S_ELEMS)           // column-major last values [320][8000]
#define L_ELEMS   (WIN * EE * NN)             //  2,560,000
#define G_OFF     (L_OFF + L_ELEMS)           // GEMM result [N*40][16]
#define G_ELEMS   (NN * WIN * 16)             //  5,120,000
// total = 33,280,000 floats = 133.1 MB

typedef __attribute__((ext_vector_type(2))) float v2f;
typedef __attribute__((ext_vector_type(8))) float v8f;

#if defined(__HIP_DEVICE_COMPILE__)
#if !__has_builtin(__builtin_amdgcn_wmma_f32_16x16x4_f32)
#error "gfx1250 WMMA f32 16x16x4 builtin not available"
#endif
#endif

// ---------------------------------------------------------------------------
// Kernel 1: per-sample sliding-window stats + pack last-values column-major.
// One block per sample n; 320 threads = (tau, e).
// Week window (5 rows) is the tail of the month window (20 rows).
// ---------------------------------------------------------------------------
__global__ __launch_bounds__(320)
void stats_pack_kernel(const float* __restrict__ x,
                       float* __restrict__ S,
                       float* __restrict__ Lcol) {
    __shared__ float xs[TT * EE];   // x[n, t, 0..7]
    const int n   = blockIdx.x;
    const int tid = threadIdx.x;

    for (int i = tid; i < TT * EE; i += 320) {
        const int t = i >> 3, e = i & 7;
        xs[i] = x[(size_t)n * (TT * FF) + t * FF + e];
    }
    __syncthreads();

    const int w = tid >> 3;   // 0..39
    const int e = tid & 7;    // 0..7

    // month window rows: w+5 .. w+24 ; week = last 5 of those
    float v[20];
#pragma unroll
    for (int j = 0; j < 20; ++j) v[j] = xs[(5 + w + j) * EE + e];

    float msum = 0.f, mmax = v[0], mmin = v[0];
#pragma unroll
    for (int j = 0; j < 20; ++j) {
        msum += v[j];
        mmax = fmaxf(mmax, v[j]);
        mmin = fminf(mmin, v[j]);
    }
    const float mmean = msum * (1.0f / 20.0f);
    float mss = 0.f;
#pragma unroll
    for (int j = 0; j < 20; ++j) { float d = v[j] - mmean; mss += d * d; }
    const float mstd = sqrtf(mss * (1.0f / 19.0f));

    float wsum = 0.f, wmax = v[15], wmin = v[15];
#pragma unroll
    for (int j = 15; j < 20; ++j) {
        wsum += v[j];
        wmax = fmaxf(wmax, v[j]);
        wmin = fminf(wmin, v[j]);
    }
    const float wmean = wsum * (1.0f / 5.0f);
    float wss = 0.f;
#pragma unroll
    for (int j = 15; j < 20; ++j) { float d = v[j] - wmean; wss += d * d; }
    const float wstd = sqrtf(wss * (1.0f / 4.0f));

    float* Sp = S + ((size_t)n * WIN + w) * SDIM;
    Sp[ 0 + e] = wmean;
    Sp[ 8 + e] = wstd;
    Sp[24 + e] = wmax;
    Sp[32 + e] = wmin;
    Sp[40 + e] = mmean;
    Sp[48 + e] = mstd;
    Sp[64 + e] = mmax;
    Sp[72 + e] = mmin;

    // last value for ranking: x[n, 24+w, e]  -> column-major [col][n]
    Lcol[(size_t)(w * EE + e) * NN + n] = xs[(24 + w) * EE + e];
}

// ---------------------------------------------------------------------------
// Kernel 2: cross-sample rank per column (stable descending argsort-argsort):
// rank[n] = (#{m: v_m > v_n} + #{m: v_m == v_n && m < n}) / N
// grid = (320 columns, 8 chunks of 1000 samples); block = 256 threads.
// LDS-tiled O(N^2) counting.
// ---------------------------------------------------------------------------
#define RANK_TILE 1024
__global__ __launch_bounds__(256)
void rank_kernel(const float* __restrict__ Lcol,
                 float* __restrict__ S) {
    __shared__ float buf[RANK_TILE];
    const int col = blockIdx.x;          // w*8+e
    const int n0  = blockIdx.y * 1000;
    const int tid = threadIdx.x;
    const float* colp = Lcol + (size_t)col * NN;

    float vloc[4];
    int   nloc[4];
    int   cnt[4] = {0, 0, 0, 0};
#pragma unroll
    for (int s = 0; s < 4; ++s) {
        const int n = n0 + tid + 256 * s;
        nloc[s] = n;
        vloc[s] = (n < n0 + 1000 && n < NN) ? colp[n] : 0.f;
    }

    for (int base = 0; base < NN; base += RANK_TILE) {
        __syncthreads();
        for (int i = tid; i < RANK_TILE; i += 256) {
            const int m = base + i;
            buf[i] = (m < NN) ? colp[m] : 0.f;
        }
        __syncthreads();
        const int limit = min(RANK_TILE, NN - base);
        for (int i = 0; i < limit; ++i) {
            const float vm = buf[i];
            const int   m  = base + i;
#pragma unroll
            for (int s = 0; s < 4; ++s) {
                cnt[s] += ((vm > vloc[s]) || (vm == vloc[s] && m < nloc[s])) ? 1 : 0;
            }
        }
    }

    const int w = col >> 3, e = col & 7;
#pragma unroll
    for (int s = 0; s < 4; ++s) {
        const int n = nloc[s];
        if (n < n0 + 1000 && n < NN) {
            const float rk = (float)cnt[s] * (1.0f / (float)NN);
            float* Sp = S + ((size_t)n * WIN + w) * SDIM;
            Sp[16 + e] = rk;   // week rank channels
            Sp[56 + e] = rk;   // month rank channels (identical)
        }
    }
}

// ---------------------------------------------------------------------------
// Kernel 3: conv-as-GEMM with fp32 WMMA.
//   G[(n*40+tau), k] = sum_c XC[n,tau,c] * w127[c,k],  K = 144, N padded to 16
// One wave (32 lanes) per 16-row M-tile; 4 waves per block.
// A staged in LDS (row stride 145 to avoid bank conflicts); B = zero-padded
// weights of channel 127 shared by the whole block.
// ---------------------------------------------------------------------------
#define A_STRIDE 145
__global__ __launch_bounds__(128)
void wmma_conv_kernel(const float* __restrict__ x,
                      const float* __restrict__ S,
                      const float* __restrict__ conv_w,
                      float* __restrict__ G) {
    __shared__ float Wl[INCH * 16];            // Wl[c*16 + j], j<5 valid
    __shared__ float Al[4][16 * A_STRIDE];

    const int tid  = threadIdx.x;
    const int wave = tid >> 5;
    const int lane = tid & 31;

    // stage B: weights of output channel 127, zero padded to 16 columns
    for (int i = tid; i < INCH * 16; i += 128) {
        const int c = i >> 4, j = i & 15;
        Wl[i] = (j < KER) ? conv_w[(size_t)127 * INCH * KER + c * KER + j] : 0.f;
    }

    const int tile = blockIdx.x * 4 + wave;    // 20000 tiles of 16 rows
    const int r0   = tile * 16;

    // stage A: 16 rows x 144 cols of XC
    float* A = Al[wave];
    for (int idx = lane; idx < 16 * INCH; idx += 32) {
        const int i = idx / INCH;
        const int c = idx - i * INCH;
        const int r = r0 + i;
        const int n = r / WIN;
        const int tau = r - n * WIN;
        float v;
        if (c < FF) v = x[(size_t)n * (TT * FF) + (24 + tau) * FF + c];
        else        v = S[((size_t)n * WIN + tau) * SDIM + (c - FF)];
        A[i * A_STRIDE + c] = v;
    }
    __syncthreads();

    // WMMA accumulate: 36 steps of K=4
    const int half = lane >> 4;       // 0: K+{0,1}, 1: K+{2,3}
    const int m    = lane & 15;       // A row (M) / B,D column (N)
    const int o    = half * 2;
    v8f acc = {};
#pragma unroll
    for (int kk = 0; kk < INCH; kk += 4) {
        v2f a = { A[m * A_STRIDE + kk + o], A[m * A_STRIDE + kk + o + 1] };
        v2f b = { Wl[(kk + o) * 16 + m],    Wl[(kk + o + 1) * 16 + m] };
        acc = __builtin_amdgcn_wmma_f32_16x16x4_f32(
                  false, a, false, b, (short)0, acc, false, false);
    }

    // D layout: VGPR v -> row M = v + half*8, col N = m
#pragma unroll
    for (int v = 0; v < 8; ++v) {
        const int rowM = v + half * 8;
        G[(size_t)(r0 + rowM) * 16 + m] = acc[v];
    }
}

// ---------------------------------------------------------------------------
// Kernel 4: epilogue. H[n,t] = sum_k G[n*40+t+k, k] + b127 ; leaky ; dot lin_w
// ---------------------------------------------------------------------------
__global__ __launch_bounds__(256)
void epilogue_kernel(const float* __restrict__ G,
                     const float* __restrict__ conv_b,
                     const float* __restrict__ lin_w,
                     const float* __restrict__ lin_b,
                     float* __restrict__ out) {
    const int n = blockIdx.x * blockDim.x + threadIdx.x;
    if (n >= NN) return;
    const float b127 = conv_b[127];
    float acc = lin_b[0];
    const float* Gn = G + (size_t)n * WIN * 16;
#pragma unroll 4
    for (int t = 0; t < LOUT; ++t) {
        float h = b127;
#pragma unroll
        for (int k = 0; k < KER; ++k) h += Gn[(t + k) * 16 + k];
        h = (h >= 0.f) ? h : ALPHA * h;
        acc += h * lin_w[t];
    }
    out[n] = acc;
}

// ---------------------------------------------------------------------------
extern "C" void kernel_launch(void* const* d_in, const int* in_sizes, int n_in,
                              void* d_out, int out_size, void* d_ws, size_t ws_size,
                              hipStream_t stream) {
    const float* x      = (const float*)d_in[0];
    const float* conv_w = (const float*)d_in[1];
    const float* conv_b = (const float*)d_in[2];
    const float* lin_w  = (const float*)d_in[3];
    const float* lin_b  = (const float*)d_in[4];
    float* out = (float*)d_out;

    float* ws   = (float*)d_ws;
    float* S    = ws + S_OFF;
    float* Lcol = ws + L_OFF;
    float* G    = ws + G_OFF;

    stats_pack_kernel<<<NN, 320, 0, stream>>>(x, S, Lcol);
    rank_kernel<<<dim3(WIN * EE, 8), 256, 0, stream>>>(Lcol, S);
    wmma_conv_kernel<<<(NN * WIN) / (16 * 4), 128, 0, stream>>>(x, S, conv_w, G);
    epilogue_kernel<<<(NN + 255) / 256, 256, 0, stream>>>(G, conv_b, lin_w, lin_b, out);
}